// TwoHotCategoricalLoss_62405874810994
// MI455X (gfx1250) — compile-verified
//
#include <hip/hip_runtime.h>
#include <hip/hip_bf16.h>

#define NCOLS 256
#define MINV (-20.0f)
#define MAXV (20.0f)
#define NBINS 255
#define BLOCKS 2048   // 16384 waves -> 16 waves/SIMD32 for latency hiding

typedef float v2f __attribute__((ext_vector_type(2)));
typedef float v8f __attribute__((ext_vector_type(8)));

// Kernel 1: one wave (32 lanes) per row. Each lane holds 8 contiguous floats
// (two b128 loads -> 1KB contiguous per wave). Row max via shuffle tree,
// cross-lane exp-sum via V_WMMA_F32_16X16X4_F32 (A = per-lane partials,
// B = ones => D[m][n] = p_m + p_{m+16}), then one shfl_xor(16).
__global__ __launch_bounds__(256) void twohot_loss_partial(
    const float* __restrict__ logits, const float* __restrict__ target,
    float* __restrict__ partials, int n_rows)
{
    const int lane = threadIdx.x & 31;
    const int wid  = threadIdx.x >> 5;                       // wave in block: 0..7
    const int wavesPerGrid = (gridDim.x * blockDim.x) >> 5;  // 16384
    const int gw = blockIdx.x * (blockDim.x >> 5) + wid;     // global wave id

    const float delta     = (MAXV - MINV) / (float)NBINS;
    const float inv_delta = (float)NBINS / (MAXV - MINV);

    float acc = 0.0f;

    for (int r = gw; r < n_rows; r += wavesPerGrid) {
        const float* row   = logits + (size_t)r * NCOLS;
        const float4* row4 = reinterpret_cast<const float4*>(row);
        float4 x0 = row4[lane * 2 + 0];
        float4 x1 = row4[lane * 2 + 1];

        // ---- row max ----
        float m = fmaxf(fmaxf(fmaxf(x0.x, x0.y), fmaxf(x0.z, x0.w)),
                        fmaxf(fmaxf(x1.x, x1.y), fmaxf(x1.z, x1.w)));
        m = fmaxf(m, __shfl_xor(m, 1));
        m = fmaxf(m, __shfl_xor(m, 2));
        m = fmaxf(m, __shfl_xor(m, 4));
        m = fmaxf(m, __shfl_xor(m, 8));
        m = fmaxf(m, __shfl_xor(m, 16));

        // ---- per-lane exp partial ----
        float p = __expf(x0.x - m) + __expf(x0.y - m) +
                  __expf(x0.z - m) + __expf(x0.w - m) +
                  __expf(x1.x - m) + __expf(x1.y - m) +
                  __expf(x1.z - m) + __expf(x1.w - m);

        // ---- cross-lane sum via WMMA f32 16x16x4 ----
        // A-matrix 16x4 f32 layout: lane n<16: v0=A[n][0], v1=A[n][1];
        // lane n+16: v0=A[n][2], v1=A[n][3]. With B = ones:
        //   D[m][n] = p_m + p_{m+16}   (replicated across N)
        v2f A; A[0] = p;    A[1] = 0.0f;
        v2f B; B[0] = 1.0f; B[1] = 1.0f;
        v8f Cm = {};
        v8f D = __builtin_amdgcn_wmma_f32_16x16x4_f32(
                    false, A, false, B, (short)0, Cm, false, false);
        // lanes 0..15 hold m=0..7 in v0..v7; lanes 16..31 hold m=8..15
        float s = ((D[0] + D[1]) + (D[2] + D[3])) +
                  ((D[4] + D[5]) + (D[6] + D[7]));
        s += __shfl_xor(s, 16);   // full 32-lane sum in every lane

        float lse = m + __logf(s);

        // ---- two-hot gather + loss ----
        float t  = target[r];
        float u  = (t - MINV) * inv_delta;
        int   k  = (int)floorf(u);
        int idx  = k + 1;
        idx = idx < 1 ? 1 : (idx > NBINS ? NBINS : idx);
        float lo   = MINV + (float)(idx - 1) * delta;
        float w_hi = (t - lo) * inv_delta;
        float w_lo = 1.0f - w_hi;
        float xlo = row[idx - 1];   // broadcast load, row is cache-hot
        float xhi = row[idx];
        float loss = lse - (w_lo * xlo + w_hi * xhi);
        if (lane == 0) acc += loss;
    }

    __shared__ float s_w[8];
    if (lane == 0) s_w[wid] = acc;
    __syncthreads();
    if (threadIdx.x == 0) {
        float t = 0.0f;
        for (int i = 0; i < 8; ++i) t += s_w[i];
        partials[blockIdx.x] = t;
    }
}

// Kernel 2: deterministic tree reduction of the BLOCKS block partials.
__global__ __launch_bounds__(1024) void reduce_partials(
    const float* __restrict__ partials, int n, float* __restrict__ out,
    float inv_n_rows)
{
    __shared__ float s[1024];
    int t = threadIdx.x;
    float v = 0.0f;
    for (int i = t; i < n; i += 1024) v += partials[i];
    s[t] = v;
    __syncthreads();
    for (int off = 512; off > 0; off >>= 1) {
        if (t < off) s[t] += s[t + off];
        __syncthreads();
    }
    if (t == 0) out[0] = s[0] * inv_n_rows;
}

extern "C" void kernel_launch(void* const* d_in, const int* in_sizes, int n_in,
                              void* d_out, int out_size, void* d_ws, size_t ws_size,
                              hipStream_t stream) {
    const float* logits = (const float*)d_in[0];
    const float* target = (const float*)d_in[1];
    float* out      = (float*)d_out;
    float* partials = (float*)d_ws;          // BLOCKS floats of scratch (8KB)
    int n_rows = in_sizes[1];                // N = 131072

    twohot_loss_partial<<<BLOCKS, 256, 0, stream>>>(logits, target, partials, n_rows);
    reduce_partials<<<1, 1024, 0, stream>>>(partials, BLOCKS, out, 1.0f / (float)n_rows);
}